// _GptOssGate_85787676770790
// MI455X (gfx1250) — compile-verified
//
#include <hip/hip_runtime.h>
#include <hip/hip_bf16.h>

#define HIDDEN   2880
#define NEXP     32
#define TOPK     4
#define TOKENS   16384
#define WAVES_PER_BLOCK 4
#define ROWS_PER_WAVE   32           // two 16-row M-tiles per wave
#define LDS_STRIDE      33           // 32 + 1 pad to avoid bank conflicts

typedef __attribute__((ext_vector_type(2))) float v2f;
typedef __attribute__((ext_vector_type(4))) float v4f;
typedef __attribute__((ext_vector_type(8))) float v8f;
typedef __attribute__((ext_vector_type(4))) int   v4i;

__global__ __launch_bounds__(32 * WAVES_PER_BLOCK)
void gptoss_gate_kernel(const float* __restrict__ hs,     // [TOKENS, HIDDEN]
                        const float* __restrict__ wgt,    // [NEXP, HIDDEN]
                        const float* __restrict__ bias,   // [NEXP]
                        float*       __restrict__ out_w,  // [TOKENS, TOPK]
                        int*         __restrict__ out_i)  // [TOKENS, TOPK]
{
    __shared__ float lds[WAVES_PER_BLOCK * ROWS_PER_WAVE * LDS_STRIDE];

    const int lane = threadIdx.x & 31;
    const int wave = threadIdx.x >> 5;
    const int tile = blockIdx.x * WAVES_PER_BLOCK + wave;
    const int row0 = tile * ROWS_PER_WAVE;       // first token of this wave

    const int m  = lane & 15;                    // A-row / B-col within a tile
    const int ko = (lane >> 4) * 2;              // K sub-offset per ISA A/B layout

    // A: two 16-row token tiles; B: expert rows 0..15 and 16..31 of weight.
    const float* __restrict__ a0ptr = hs  + (size_t)(row0 + m)      * HIDDEN + ko;
    const float* __restrict__ a1ptr = hs  + (size_t)(row0 + 16 + m) * HIDDEN + ko;
    const float* __restrict__ b0ptr = wgt + (size_t)m               * HIDDEN + ko;
    const float* __restrict__ b1ptr = wgt + (size_t)(m + 16)        * HIDDEN + ko;

    v8f acc00 = {0.f,0.f,0.f,0.f,0.f,0.f,0.f,0.f};   // Mtile0 x experts 0-15
    v8f acc01 = acc00;                                // Mtile0 x experts 16-31
    v8f acc10 = acc00;                                // Mtile1 x experts 0-15
    v8f acc11 = acc00;                                // Mtile1 x experts 16-31

    #pragma unroll 4
    for (int k = 0; k < HIDDEN; k += 4) {
        v2f a0 = *(const v2f*)(a0ptr + k);
        v2f a1 = *(const v2f*)(a1ptr + k);
        v2f b0 = *(const v2f*)(b0ptr + k);
        v2f b1 = *(const v2f*)(b1ptr + k);
        // Exact fp32 on the matrix pipe; each load feeds two WMMAs.
        acc00 = __builtin_amdgcn_wmma_f32_16x16x4_f32(false, a0, false, b0,
                                                      (short)0, acc00, false, false);
        acc01 = __builtin_amdgcn_wmma_f32_16x16x4_f32(false, a0, false, b1,
                                                      (short)0, acc01, false, false);
        acc10 = __builtin_amdgcn_wmma_f32_16x16x4_f32(false, a1, false, b0,
                                                      (short)0, acc10, false, false);
        acc11 = __builtin_amdgcn_wmma_f32_16x16x4_f32(false, a1, false, b1,
                                                      (short)0, acc11, false, false);
    }

    // Bias is constant per lane in the C/D layout (N == lane&15 in every VGPR).
    const float bb0 = bias[m];
    const float bb1 = bias[m + 16];

    // Scatter logits to LDS. For each M-tile: VGPR j of lanes 0-15 is row j,
    // lanes 16-31 row j+8; M-tile 1 adds 16 rows.
    float* myLds = lds + wave * (ROWS_PER_WAVE * LDS_STRIDE);
    const int rbase = (lane >> 4) * 8;
    #pragma unroll
    for (int j = 0; j < 8; ++j) {
        myLds[(rbase + j) * LDS_STRIDE + m]           = acc00[j] + bb0;
        myLds[(rbase + j) * LDS_STRIDE + m + 16]      = acc01[j] + bb1;
        myLds[(rbase + j + 16) * LDS_STRIDE + m]      = acc10[j] + bb0;
        myLds[(rbase + j + 16) * LDS_STRIDE + m + 16] = acc11[j] + bb1;
    }
    asm volatile("s_wait_dscnt 0" ::: "memory");

    // Full wave: one token per lane -> serial top-4 + softmax over the 4 picks.
    {
        const float* rowp = myLds + lane * LDS_STRIDE;
        float bv0 = -3.402823e38f, bv1 = bv0, bv2 = bv0, bv3 = bv0;
        int   bi0 = 0, bi1 = 0, bi2 = 0, bi3 = 0;
        #pragma unroll
        for (int e = 0; e < NEXP; ++e) {
            float v = rowp[e];
            if (v > bv3) {
                if (v > bv0) {
                    bv3 = bv2; bi3 = bi2; bv2 = bv1; bi2 = bi1;
                    bv1 = bv0; bi1 = bi0; bv0 = v;   bi0 = e;
                } else if (v > bv1) {
                    bv3 = bv2; bi3 = bi2; bv2 = bv1; bi2 = bi1;
                    bv1 = v;   bi1 = e;
                } else if (v > bv2) {
                    bv3 = bv2; bi3 = bi2; bv2 = v;   bi2 = e;
                } else {
                    bv3 = v;   bi3 = e;
                }
            }
        }
        float e0 = 1.0f;                    // exp(bv0 - bv0)
        float e1 = __expf(bv1 - bv0);
        float e2 = __expf(bv2 - bv0);
        float e3 = __expf(bv3 - bv0);
        float inv = 1.0f / (e0 + e1 + e2 + e3);

        const int tok = row0 + lane;
        v4f wv = { e0 * inv, e1 * inv, e2 * inv, e3 * inv };
        v4i iv = { bi0, bi1, bi2, bi3 };
        *(v4f*)(out_w + (size_t)tok * TOPK) = wv;
        *(v4i*)(out_i + (size_t)tok * TOPK) = iv;
    }
}

extern "C" void kernel_launch(void* const* d_in, const int* in_sizes, int n_in,
                              void* d_out, int out_size, void* d_ws, size_t ws_size,
                              hipStream_t stream) {
    (void)in_sizes; (void)n_in; (void)out_size; (void)d_ws; (void)ws_size;

    const float* hs   = (const float*)d_in[0];   // [TOKENS, HIDDEN] f32
    const float* wgt  = (const float*)d_in[1];   // [NEXP, HIDDEN]   f32
    const float* bias = (const float*)d_in[2];   // [NEXP]           f32

    float* out_w = (float*)d_out;                         // routing_weights
    int*   out_i = ((int*)d_out) + (size_t)TOKENS * TOPK; // selected_experts

    const int tiles  = TOKENS / ROWS_PER_WAVE;            // 512 waves
    const int blocks = tiles / WAVES_PER_BLOCK;           // 128 blocks
    gptoss_gate_kernel<<<blocks, 32 * WAVES_PER_BLOCK, 0, stream>>>(
        hs, wgt, bias, out_w, out_i);
}